// DiffVAE_62508954026002
// MI455X (gfx1250) — compile-verified
//
#include <hip/hip_runtime.h>
#include <hip/hip_bf16.h>
#include <math.h>

// Problem constants (fixed by the reference)
#define B_  32
#define L_  40
#define H_  448
#define S_  8000
#define NC_ 25
#define T_  (S_ * NC_)

typedef float v2f __attribute__((ext_vector_type(2)));
typedef float v8f __attribute__((ext_vector_type(8)));

// ---------------------------------------------------------------------------
// Kernel 0: zero the two accumulators in workspace
// ---------------------------------------------------------------------------
__global__ void k_init(float* __restrict__ acc) {
    if (threadIdx.x < 2) acc[threadIdx.x] = 0.0f;
}

// ---------------------------------------------------------------------------
// Kernel 1: x_sum[b,h] = sum_l x[b,l,h]   (32x448 outputs)
// ---------------------------------------------------------------------------
__global__ void k_lsum(const float* __restrict__ x, float* __restrict__ xsum) {
    int i = blockIdx.x * blockDim.x + threadIdx.x;
    if (i >= B_ * H_) return;
    int b = i / H_;
    int h = i - b * H_;
    const float* p = x + (size_t)b * L_ * H_ + h;
    float s = 0.0f;
#pragma unroll
    for (int l = 0; l < L_; ++l) s += p[(size_t)l * H_];
    xsum[i] = s;
}

// ---------------------------------------------------------------------------
// Kernel 2: x_pool[b,n] = sum_k x_sum[b,k] * W[n,k]
// WMMA f32 16x16x4: A = x_sum tile (16x4), B = W^T tile (4x16), C/D f32 16x16.
// 56 output tiles (2 M-tiles x 28 N-tiles), one wave per tile, 112 WMMAs/wave.
// ---------------------------------------------------------------------------
__global__ void k_gemm(const float* __restrict__ xsum,
                       const float* __restrict__ W,
                       float* __restrict__ xpool) {
    int wave = (blockIdx.x * blockDim.x + threadIdx.x) >> 5;
    int lane = threadIdx.x & 31;
    if (wave >= (B_ / 16) * (H_ / 16)) return;
    int mt = wave / (H_ / 16);
    int nt = wave - mt * (H_ / 16);

    int mrow  = mt * 16 + (lane & 15);     // A: lane holds row M = lane&15
    int ncol  = nt * 16 + (lane & 15);     // B: lane holds col N = lane&15
    int khalf = (lane >> 4) * 2;           // lanes 16-31 hold K+2, K+3

    v8f c = {0.f, 0.f, 0.f, 0.f, 0.f, 0.f, 0.f, 0.f};

#if defined(__has_builtin) && __has_builtin(__builtin_amdgcn_wmma_f32_16x16x4_f32)
    for (int k = 0; k < H_; k += 4) {
        const float* ap = xsum + (size_t)mrow * H_ + k + khalf;
        const float* bp = W    + (size_t)ncol * H_ + k + khalf;  // B[k,n] = W[n,k]
        v2f a; a.x = ap[0]; a.y = ap[1];
        v2f b; b.x = bp[0]; b.y = bp[1];
        c = __builtin_amdgcn_wmma_f32_16x16x4_f32(
                /*neg_a=*/false, a, /*neg_b=*/false, b,
                /*c_mod=*/(short)0, c, /*reuse_a=*/false, /*reuse_b=*/false);
    }
#else
    // Scalar fallback honoring the 16x16 f32 C/D VGPR layout
    for (int j = 0; j < 8; ++j) {
        int row = mt * 16 + j + ((lane >> 4) << 3);
        float s = 0.0f;
        for (int k = 0; k < H_; ++k)
            s += xsum[(size_t)row * H_ + k] * W[(size_t)ncol * H_ + k];
        c[j] = s;
    }
#endif

    // D layout: VGPR j -> M = j (lanes 0-15), M = j+8 (lanes 16-31); N = lane&15
    int rbase = mt * 16 + ((lane >> 4) << 3);
#pragma unroll
    for (int j = 0; j < 8; ++j)
        xpool[(size_t)(rbase + j) * H_ + ncol] = c[j];
}

// ---------------------------------------------------------------------------
// Kernel 3: streaming scores + segment softmax-CE.
// x_pool (57 KB) staged in LDS; one wave per 25-candidate segment.
// cand_vecs streams from HBM once (358 MB -> ~15 us roofline at 23.3 TB/s).
// ---------------------------------------------------------------------------
__global__ void __launch_bounds__(256)
k_score(const float* __restrict__ cand,
        const float* __restrict__ xpool,
        const int*   __restrict__ bidx,
        const int*   __restrict__ lab,
        float*       __restrict__ acc) {
    __shared__ float xls[B_ * H_];  // 57344 B

    // Cooperative LDS fill with b128 copies
    const float4* src = (const float4*)xpool;
    float4*       dst = (float4*)xls;
    for (int i = threadIdx.x; i < (B_ * H_) / 4; i += blockDim.x) dst[i] = src[i];
    __syncthreads();

    int wave = threadIdx.x >> 5;
    int lane = threadIdx.x & 31;
    int seg  = blockIdx.x * (blockDim.x >> 5) + wave;
    if (seg >= S_) return;

    int   li     = lab[seg];
    float m      = -INFINITY;
    float ssum   = 0.0f;
    float slabel = 0.0f;

    for (int j = 0; j < NC_; ++j) {
        int t = seg * NC_ + j;
        int b = bidx[t];
        const float4* cp  = (const float4*)(cand + (size_t)t * H_);
        const float4* xq  = (const float4*)(xls  + b * H_);
        float dot = 0.0f;
#pragma unroll
        for (int i = 0; i < 3; ++i) {       // 3 x (32 lanes x float4) = 384 floats
            float4 cv = cp[lane + 32 * i];
            float4 xv = xq[lane + 32 * i];
            dot += cv.x * xv.x + cv.y * xv.y + cv.z * xv.z + cv.w * xv.w;
        }
        {                                    // tail 64 floats: 32 lanes x float2
            const float2* cp2 = (const float2*)(cand + (size_t)t * H_ + 384);
            const float2* xq2 = (const float2*)(xls  + b * H_ + 384);
            float2 cv = cp2[lane];
            float2 xv = xq2[lane];
            dot += cv.x * xv.x + cv.y * xv.y;
        }
        // wave32 butterfly reduction -> all lanes hold the full dot
#pragma unroll
        for (int off = 16; off >= 1; off >>= 1)
            dot += __shfl_xor(dot, off, 32);

        // online (max, sum-exp) update, uniform across lanes
        float m2 = fmaxf(m, dot);
        ssum = ssum * __expf(m - m2) + __expf(dot - m2);
        m    = m2;
        if (j == li) slabel = dot;
    }

    if (lane == 0) {
        float lse = m + __logf(ssum);
        atomicAdd(&acc[0], lse - slabel);
        atomicAdd(&acc[1], (slabel >= m) ? 1.0f : 0.0f);
    }
}

// ---------------------------------------------------------------------------
// Kernel 4: finalize -> d_out = [loss, acc]
// ---------------------------------------------------------------------------
__global__ void k_fin(const float* __restrict__ acc, float* __restrict__ out) {
    if (threadIdx.x == 0) {
        out[0] = acc[0] / (float)B_;
        out[1] = acc[1] / (float)S_;
    }
}

// ---------------------------------------------------------------------------
extern "C" void kernel_launch(void* const* d_in, const int* in_sizes, int n_in,
                              void* d_out, int out_size, void* d_ws, size_t ws_size,
                              hipStream_t stream) {
    const float* x    = (const float*)d_in[0];   // [B,L,H]
    const float* cand = (const float*)d_in[1];   // [T,H]
    const float* W    = (const float*)d_in[2];   // [H,H]
    const int*   bidx = (const int*)d_in[3];     // [T]
    const int*   lab  = (const int*)d_in[4];     // [S]

    float* ws    = (float*)d_ws;
    float* xsum  = ws;                 // 32*448 floats
    float* xpool = ws + B_ * H_;       // 32*448 floats
    float* acc   = ws + 2 * B_ * H_;   // 2 floats

    k_init<<<1, 32, 0, stream>>>(acc);
    k_lsum<<<(B_ * H_ + 255) / 256, 256, 0, stream>>>(x, xsum);
    k_gemm<<<14, 128, 0, stream>>>(xsum, W, xpool);          // 56 waves, 56 tiles
    k_score<<<S_ / 8, 256, 0, stream>>>(cand, xpool, bidx, lab, acc);
    k_fin<<<1, 32, 0, stream>>>(acc, (float*)d_out);
}